// QwenMoeLayerGather_43104291782789
// MI455X (gfx1250) — compile-verified
//
#include <hip/hip_runtime.h>

typedef float v2f __attribute__((ext_vector_type(2)));
typedef float v8f __attribute__((ext_vector_type(8)));

#define HIDDEN 2048
#define INTER  1408
#define TOPK   4

// ---------------------------------------------------------------------------
// Phase 1: gate & up GEMVs + silu-combine.
// grid = (INTER/16, TOPK), block = 256 (8 waves).
// waves 0-3: gate, K-chunks of 512.  waves 4-7: up, K-chunks of 512.
// Each wave owns the same 16 output rows; WMMA f32 16x16x4 accumulates the
// dot products (B operand = x chunk broadcast across all 16 columns).
// ---------------------------------------------------------------------------
__global__ __launch_bounds__(256) void moe_gate_up(
    const float* __restrict__ x,
    const int*   __restrict__ topk_idx,
    const float* __restrict__ gate_all,
    const float* __restrict__ up_all,
    float*       __restrict__ inter_ws)
{
    __shared__ float xs[HIDDEN];
    __shared__ float red[8][16];

    const int tid = threadIdx.x;

    // Stage x (8 KB) into LDS: 256 threads x 4 float2 loads.
    {
        const v2f* src = (const v2f*)x;
        v2f*       dst = (v2f*)xs;
#pragma unroll
        for (int i = 0; i < (HIDDEN / 2) / 256; ++i)
            dst[tid + i * 256] = src[tid + i * 256];
    }
    __syncthreads();

    const int wave = tid >> 5;
    const int lane = tid & 31;
    const int half = lane >> 4;       // 0: K=0,1   1: K=2,3
    const int rowl = lane & 15;       // M within the 16-row tile
    const int koff = half * 2;

    const int kexp = blockIdx.y;                 // which of the top-k slots
    const int e    = topk_idx[kexp];             // expert id
    const int r0   = blockIdx.x * 16;            // output-row base

    const int mtx   = wave >> 2;                 // 0 = gate, 1 = up
    const int chunk = wave & 3;
    const int k0    = chunk * (HIDDEN / 4);      // 512-wide K chunk

    const float* W = (mtx ? up_all : gate_all)
                   + (size_t)e * (size_t)(INTER * HIDDEN)
                   + (size_t)(r0 + rowl) * HIDDEN + koff;
    const float* xb = xs + koff;

    v8f acc = {};
#pragma unroll 4
    for (int kb = k0; kb < k0 + HIDDEN / 4; kb += 4) {
        v2f a = *(const v2f*)(W + kb);    // 16x4 weight tile (A operand)
        v2f b = *(const v2f*)(xb + kb);   // x chunk broadcast (B operand)
        acc = __builtin_amdgcn_wmma_f32_16x16x4_f32(
            false, a, false, b, (short)0, acc, false, false);
    }

    // D layout: vgpr v -> M=v (lanes 0-15), M=v+8 (lanes 16-31); all N equal.
    if (lane == 0) {
#pragma unroll
        for (int v = 0; v < 8; ++v) red[wave][v] = acc[v];
    } else if (lane == 16) {
#pragma unroll
        for (int v = 0; v < 8; ++v) red[wave][8 + v] = acc[v];
    }
    __syncthreads();

    if (tid < 16) {
        float g = red[0][tid] + red[1][tid] + red[2][tid] + red[3][tid];
        float u = red[4][tid] + red[5][tid] + red[6][tid] + red[7][tid];
        float s = g / (1.0f + __expf(-g));        // silu(g)
        inter_ws[kexp * INTER + r0 + tid] = s * u;
    }
}

// ---------------------------------------------------------------------------
// Phase 2: down GEMV + weighted sum over experts.
// grid = HIDDEN/16, block = 256 (8 waves): wave w -> expert w>>1, K-half w&1.
// ---------------------------------------------------------------------------
__global__ __launch_bounds__(256) void moe_down(
    const float* __restrict__ inter_ws,
    const int*   __restrict__ topk_idx,
    const float* __restrict__ topk_w,
    const float* __restrict__ down_all,
    float*       __restrict__ out)
{
    __shared__ float is[TOPK * INTER];
    __shared__ float red[8][16];

    const int tid = threadIdx.x;

    // Stage all 4 intermediate vectors (22.5 KB) into LDS.
    {
        const v2f* src = (const v2f*)inter_ws;
        v2f*       dst = (v2f*)is;
#pragma unroll
        for (int i = 0; i < (TOPK * INTER / 2) / 256; ++i)   // 11 iters exact
            dst[tid + i * 256] = src[tid + i * 256];
    }
    __syncthreads();

    const int wave = tid >> 5;
    const int lane = tid & 31;
    const int half = lane >> 4;
    const int rowl = lane & 15;
    const int koff = half * 2;

    const int kexp  = wave >> 1;
    const int khalf = wave & 1;
    const int k0    = khalf * (INTER / 2);       // 704-wide K chunk
    const int r0    = blockIdx.x * 16;
    const int e     = topk_idx[kexp];

    const float* W = down_all
                   + (size_t)e * (size_t)(HIDDEN * INTER)
                   + (size_t)(r0 + rowl) * INTER + koff;
    const float* xb = is + kexp * INTER + koff;

    v8f acc = {};
#pragma unroll 4
    for (int kb = k0; kb < k0 + INTER / 2; kb += 4) {
        v2f a = *(const v2f*)(W + kb);
        v2f b = *(const v2f*)(xb + kb);
        acc = __builtin_amdgcn_wmma_f32_16x16x4_f32(
            false, a, false, b, (short)0, acc, false, false);
    }

    const float wgt = topk_w[kexp];
    if (lane == 0) {
#pragma unroll
        for (int v = 0; v < 8; ++v) red[wave][v] = acc[v] * wgt;
    } else if (lane == 16) {
#pragma unroll
        for (int v = 0; v < 8; ++v) red[wave][8 + v] = acc[v] * wgt;
    }
    __syncthreads();

    if (tid < 16) {
        float s = 0.0f;
#pragma unroll
        for (int w = 0; w < 8; ++w) s += red[w][tid];
        out[r0 + tid] = s;
    }
}

// ---------------------------------------------------------------------------
extern "C" void kernel_launch(void* const* d_in, const int* in_sizes, int n_in,
                              void* d_out, int out_size, void* d_ws, size_t ws_size,
                              hipStream_t stream)
{
    const float* x        = (const float*)d_in[0];
    const int*   topk_idx = (const int*)  d_in[1];
    const float* topk_w   = (const float*)d_in[2];
    const float* gate_all = (const float*)d_in[3];
    const float* up_all   = (const float*)d_in[4];
    const float* down_all = (const float*)d_in[5];
    float*       out      = (float*)d_out;
    float*       ws       = (float*)d_ws;   // TOPK*INTER floats = 22.5 KB

    dim3 g1(INTER / 16, TOPK);               // 88 x 4 blocks
    moe_gate_up<<<g1, 256, 0, stream>>>(x, topk_idx, gate_all, up_all, ws);

    moe_down<<<HIDDEN / 16, 256, 0, stream>>>(ws, topk_idx, topk_w, down_all, out);
}